// EquivariantMLP_68745246540041
// MI455X (gfx1250) — compile-verified
//
#include <hip/hip_runtime.h>

// ---------------------------------------------------------------------------
// EquivariantMLP fused kernel for gfx1250 (CDNA5, wave32, WMMA).
// V_WMMA_F32_16X16X4_F32 for all matmuls; h ping-pongs in LDS; fast-rcp SiLU;
// wave-uniform control flow via readfirstlane (no EXEC divergence).
// ---------------------------------------------------------------------------

typedef float v2f __attribute__((ext_vector_type(2)));
typedef float v8f __attribute__((ext_vector_type(8)));

#define NUM_H   64
#define DIN     1024
#define HPITCH  1026      // padded row pitch (dwords): even (b64-aligned), !=0 mod 64
#define GPITCH  196       // gate buffer pitch
#define NORM    0.125f    // 1/sqrt(64)
#define CSTV    1.6768f   // 1/sqrt(E[silu(z)^2]), z~N(0,1)

__device__ __forceinline__ float silu_n(float x) {
    // CST * x * sigmoid(x), with fast v_rcp_f32 instead of IEEE divide
    return (CSTV * x) * __builtin_amdgcn_rcpf(1.0f + __expf(-x));
}

__device__ __forceinline__ v8f wmma4(v2f a, v2f b, v8f c) {
    // 8 args: (neg_a, A, neg_b, B, c_mod, C, reuse_a, reuse_b)
    return __builtin_amdgcn_wmma_f32_16x16x4_f32(false, a, false, b, (short)0, c,
                                                 false, false);
}

// One e3nn Linear+Gate layer on a 16-atom tile held in LDS.
// P: input features (16 x 1024, pitch HPITCH). Q: output. gbuf: gate scratch.
__device__ __forceinline__ void run_layer(
    const float* P, float* Q, float* gbuf,
    const float* __restrict__ W0, const float* __restrict__ W1,
    const float* __restrict__ W2, const float* __restrict__ W3,
    int w, int wu, int ln, int hi)
{
    // ---- stage 1: y0 = x0 @ W0 (64 x 256); silu; s -> Q[:, :64], g -> gbuf
    {
        v8f acc[4] = {};
        for (int k0 = 0; k0 < 64; k0 += 4) {
            const int k = k0 + 2 * hi;          // this lane's K pair base
            v2f a;
            a.x = P[ln * HPITCH + k];
            a.y = P[ln * HPITCH + k + 1];
#pragma unroll
            for (int j = 0; j < 4; ++j) {
                const int n = (4 * w + j) * 16 + ln;
                v2f b;
                b.x = W0[k * 256 + n];
                b.y = W0[(k + 1) * 256 + n];
                acc[j] = wmma4(a, b, acc[j]);
            }
        }
        if (wu == 0) {                    // scalar-uniform: wave 0 owns cols 0..63 (s)
#pragma unroll
            for (int j = 0; j < 4; ++j) {
                const int col = j * 16 + ln;
#pragma unroll
                for (int r = 0; r < 8; ++r) {
                    const int row = r + 8 * hi;
                    Q[row * HPITCH + col] = silu_n(acc[j][r] * NORM);
                }
            }
        } else {                          // waves 1..3 own cols 64..255 (g)
#pragma unroll
            for (int j = 0; j < 4; ++j) {
                const int gcol = (4 * w + j) * 16 + ln - NUM_H;
#pragma unroll
                for (int r = 0; r < 8; ++r) {
                    const int row = r + 8 * hi;
                    gbuf[row * GPITCH + gcol] = silu_n(acc[j][r] * NORM);
                }
            }
        }
    }
    __syncthreads();

    // ---- stage 2: per (l, c): y = x_{l,c} @ W_l; gate; write to Q
    for (int comp = wu; comp < 15; comp += 4) {   // scalar-uniform job loop
        int l, c, off, d;
        if (comp < 3)      { l = 1; c = comp;     off = 64;  d = 3; }
        else if (comp < 8) { l = 2; c = comp - 3; off = 256; d = 5; }
        else               { l = 3; c = comp - 8; off = 576; d = 7; }
        const float* Wl = (l == 1) ? W1 : ((l == 2) ? W2 : W3);

        v8f acc[4] = {};
        for (int k0 = 0; k0 < 64; k0 += 4) {
            const int k  = k0 + 2 * hi;
            const int ab = ln * HPITCH + off + k * d + c;
            v2f a;
            a.x = P[ab];
            a.y = P[ab + d];
#pragma unroll
            for (int j = 0; j < 4; ++j) {
                const int n = j * 16 + ln;
                v2f b;
                b.x = Wl[k * 64 + n];
                b.y = Wl[(k + 1) * 64 + n];
                acc[j] = wmma4(a, b, acc[j]);
            }
        }
#pragma unroll
        for (int j = 0; j < 4; ++j) {
            const int o = j * 16 + ln;
#pragma unroll
            for (int r = 0; r < 8; ++r) {
                const int row = r + 8 * hi;
                const float g = gbuf[row * GPITCH + (l - 1) * 64 + o];
                Q[row * HPITCH + off + o * d + c] = g * (acc[j][r] * NORM);
            }
        }
    }
    __syncthreads();
}

__global__ __launch_bounds__(128)
void eqmlp_fused(const float* __restrict__ x, const int* __restrict__ bidx,
                 const float* __restrict__ W00, const float* __restrict__ W01,
                 const float* __restrict__ W02, const float* __restrict__ W03,
                 const float* __restrict__ W10, const float* __restrict__ W11,
                 const float* __restrict__ W12, const float* __restrict__ W13,
                 const float* __restrict__ Wout, float* __restrict__ out)
{
    __shared__ float bufA[16 * HPITCH];   // 64.1 KB
    __shared__ float bufB[16 * HPITCH];   // 64.1 KB
    __shared__ float gbuf[16 * GPITCH];   // 12.3 KB   (total ~140.5 KB < 320 KB)

    const int tid  = threadIdx.x;
    const int lane = tid & 31;
    const int w    = tid >> 5;                             // wave id 0..3
    const int wu   = __builtin_amdgcn_readfirstlane(w);    // scalar wave id
    const int ln   = lane & 15;
    const int hi   = lane >> 4;
    const long long base = (long long)blockIdx.x * 16;     // first atom of tile

    // cooperative load of the 16x1024 input tile (b128 global loads)
    const float* xt = x + base * DIN;
    for (int i = tid; i < 16 * DIN / 4; i += 128) {
        const int row = i >> 8;            // 256 float4 per row
        const int c4  = i & 255;
        const float4 v = reinterpret_cast<const float4*>(xt + (long long)row * DIN)[c4];
        float* dst = &bufA[row * HPITCH + 4 * c4];
        dst[0] = v.x; dst[1] = v.y; dst[2] = v.z; dst[3] = v.w;
    }
    __syncthreads();

    run_layer(bufA, bufB, gbuf, W00, W01, W02, W03, w, wu, ln, hi);
    run_layer(bufB, bufA, gbuf, W10, W11, W12, W13, w, wu, ln, hi);

    // ---- output: h[:, :64] @ Wout (64 x 128) * NORM, atomic segment-sum
    {
        v8f acc[2] = {};
        for (int k0 = 0; k0 < 64; k0 += 4) {
            const int k = k0 + 2 * hi;
            v2f a;
            a.x = bufA[ln * HPITCH + k];
            a.y = bufA[ln * HPITCH + k + 1];
#pragma unroll
            for (int j = 0; j < 2; ++j) {
                const int n = (w + 4 * j) * 16 + ln;
                v2f b;
                b.x = Wout[k * 128 + n];
                b.y = Wout[(k + 1) * 128 + n];
                acc[j] = wmma4(a, b, acc[j]);
            }
        }
#pragma unroll
        for (int j = 0; j < 2; ++j) {
            const int n = (w + 4 * j) * 16 + ln;
#pragma unroll
            for (int r = 0; r < 8; ++r) {
                const int row = r + 8 * hi;
                const int bi  = bidx[base + row];
                atomicAdd(&out[bi * 128 + n], acc[j][r] * NORM);
            }
        }
    }
}

__global__ void zero_out_kernel(float* __restrict__ out, int n) {
    const int i = blockIdx.x * blockDim.x + threadIdx.x;
    if (i < n) out[i] = 0.0f;
}

extern "C" void kernel_launch(void* const* d_in, const int* in_sizes, int n_in,
                              void* d_out, int out_size, void* d_ws, size_t ws_size,
                              hipStream_t stream) {
    (void)in_sizes; (void)n_in; (void)d_ws; (void)ws_size;
    const float* x    = (const float*)d_in[0];
    const int*   bidx = (const int*)d_in[1];
    // d_in[2] = batch_size (fixed at 512)
    const float* W00  = (const float*)d_in[3];
    const float* W01  = (const float*)d_in[4];
    const float* W02  = (const float*)d_in[5];
    const float* W03  = (const float*)d_in[6];
    const float* W10  = (const float*)d_in[7];
    const float* W11  = (const float*)d_in[8];
    const float* W12  = (const float*)d_in[9];
    const float* W13  = (const float*)d_in[10];
    const float* Wout = (const float*)d_in[11];
    float* out = (float*)d_out;

    // d_out is poisoned by the harness: zero it before the atomic segment-sum.
    zero_out_kernel<<<(out_size + 255) / 256, 256, 0, stream>>>(out, out_size);

    const int n_tiles = 131072 / 16;   // 8192 workgroups, 16 atoms each
    eqmlp_fused<<<n_tiles, 128, 0, stream>>>(x, bidx, W00, W01, W02, W03,
                                             W10, W11, W12, W13, Wout, out);
}